// DeepMarchingCube_29944511988433
// MI455X (gfx1250) — compile-verified
//
#include <hip/hip_runtime.h>
#include <hip/hip_bf16.h>
#include <math.h>

// ---------------------------------------------------------------------------
// CDNA5 / gfx1250 WMMA types
// ---------------------------------------------------------------------------
typedef __attribute__((ext_vector_type(2))) float v2f;
typedef __attribute__((ext_vector_type(8))) float v8f;

#define BATCH 2
#define S33 35937LL   // 33^3
#define S37 50653LL   // 37^3
#define S18 5832LL    // 18^3
#define S9  729LL     // 9^3
#define S4  64LL      // 4^3

// ---------------------------------------------------------------------------
// Utility: zero fill
// ---------------------------------------------------------------------------
__global__ void k_fill0(float* p, long long n) {
  long long i = (long long)blockIdx.x * blockDim.x + threadIdx.x;
  if (i < n) p[i] = 0.0f;
}

// ---------------------------------------------------------------------------
// PointNet MLP: z = (relu(x @ w1^T + b1)) @ w2^T + b2
//   x: (npts, 3), w1: (256,3), w2: (16,256), z: (npts,16)
// One wave per 16-point tile. GEMM1 via 16 WMMA f32 16x16x4 (K=4 pads K=3),
// relu into LDS, GEMM2 via 64 chained WMMAs (K=256).
// blockDim = 64 (2 waves/block).
// ---------------------------------------------------------------------------
__global__ void k_pointnet(const float* __restrict__ pts,
                           const float* __restrict__ w1, const float* __restrict__ b1,
                           const float* __restrict__ w2, const float* __restrict__ b2,
                           float* __restrict__ z, int npts) {
  __shared__ float h[2][16][257];  // padded: stride 257 avoids 64-bank conflicts
  const int lane = threadIdx.x & 31;
  const int wid  = threadIdx.x >> 5;
  const long long m0 = ((long long)blockIdx.x * 2 + wid) * 16;
  const bool active = m0 < npts;

  const int mRow = lane & 15;           // A-fragment row (both lane halves)
  const int nCol = lane & 15;           // B/C/D column
  const int kHi  = (lane >= 16) ? 2 : 0;
  const long long m = m0 + mRow;

  // ---- GEMM1: hidden[16][256] = relu(x[16][3] @ w1^T[3][256] + b1) ----
  v2f ax;
  ax.x = (active && m < npts)                ? pts[m * 3 + kHi]     : 0.0f; // K=kHi
  ax.y = (active && m < npts && kHi + 1 < 3) ? pts[m * 3 + kHi + 1] : 0.0f; // K=3 -> 0

#pragma unroll
  for (int nt = 0; nt < 16; ++nt) {
    const int h0 = nt * 16 + nCol;  // hidden unit index 0..255
    v2f bf;
    bf.x = w1[h0 * 3 + kHi];                         // K = kHi (0 or 2, <3)
    bf.y = (kHi + 1 < 3) ? w1[h0 * 3 + kHi + 1] : 0; // K = kHi+1
    v8f c = {};
    c = __builtin_amdgcn_wmma_f32_16x16x4_f32(false, ax, false, bf,
                                              (short)0, c, false, false);
    const float bb = b1[h0];
#pragma unroll
    for (int r = 0; r < 8; ++r) {
      const int mm = r + ((lane >= 16) ? 8 : 0);
      float v = c[r] + bb;
      h[wid][mm][h0] = v > 0.0f ? v : 0.0f;
    }
  }
  __syncthreads();

  // ---- GEMM2: z[16][16] = hidden[16][256] @ w2^T[256][16] + b2 ----
  v8f acc = {};
  for (int kk = 0; kk < 256; kk += 4) {
    v2f a, bf;
    a.x  = h[wid][mRow][kk + kHi];
    a.y  = h[wid][mRow][kk + kHi + 1];
    bf.x = w2[nCol * 256 + kk + kHi];
    bf.y = w2[nCol * 256 + kk + kHi + 1];
    acc = __builtin_amdgcn_wmma_f32_16x16x4_f32(false, a, false, bf,
                                                (short)0, acc, false, false);
  }
  const float b2v = b2[nCol];
#pragma unroll
  for (int r = 0; r < 8; ++r) {
    const long long mm = m0 + r + ((lane >= 16) ? 8 : 0);
    if (active && mm < npts) z[mm * 16 + nCol] = acc[r] + b2v;
  }
}

// ---------------------------------------------------------------------------
// point_to_cell: scatter-max of z into (B,16,33,33,33) grid (pre-zeroed).
// Grid init = 0 => only positive values can win => int-punned atomicMax valid.
// ---------------------------------------------------------------------------
__global__ void k_scatter(const float* __restrict__ pts, const float* __restrict__ z,
                          float* __restrict__ grid, int npts, int ptsPerB) {
  long long i = (long long)blockIdx.x * blockDim.x + threadIdx.x;
  if (i >= (long long)npts * 16) return;
  const long long p = i >> 4;
  const int c = (int)(i & 15);
  const int b = (int)(p / ptsPerB);
  float fx = pts[p * 3 + 0], fy = pts[p * 3 + 1], fz = pts[p * 3 + 2];
  int ix = (int)floorf(fx); ix = ix < 0 ? 0 : (ix > 31 ? 31 : ix);
  int iy = (int)floorf(fy); iy = iy < 0 ? 0 : (iy > 31 ? 31 : iy);
  int iz = (int)floorf(fz); iz = iz < 0 ? 0 : (iz > 31 ? 31 : iz);
  const float v = z[p * 16 + c];
  if (v > 0.0f) {
    long long idx = ((((long long)(b * 16 + c) * 33 + ix) * 33 + iy) * 33 + iz);
    atomicMax((int*)grid + idx, __float_as_int(v));
  }
}

// ---------------------------------------------------------------------------
// Implicit-GEMM 3D conv via WMMA f32 16x16x4.
//   in : (B, Cin, Xi,Xi,Xi)    w(conv):  (Cout, Cin, 27)
//   out: (B, Cout,Xo,Xo,Xo)    w(convT): (Cin, Cout, 27), taps flipped
//   pad may be negative (crop).  Xo = Xi + 2*pad - 2, ix = ox + dx - pad.
// One wave computes one 16-voxel x 16-Cout tile. K = 27 taps x Cin.
// Tile decomposition: waves in a block share the SAME Cout tile (uniform
// weight stream -> L0 broadcast) and take CONSECUTIVE voxel tiles (A loads
// of neighbors share cachelines). blockDim = 256 (8 waves).
// ---------------------------------------------------------------------------
__global__ void k_conv_wmma(const float* __restrict__ in, const float* __restrict__ w,
                            const float* __restrict__ bias, float* __restrict__ out,
                            int Cin, int Cout, int Xi, int Xo, int pad, int flipT) {
  const int lane = threadIdx.x & 31;
  const int wid  = threadIdx.x >> 5;
  const long long Si = (long long)Xi * Xi * Xi;
  const long long So = (long long)Xo * Xo * Xo;
  const long long M  = (long long)BATCH * So;
  const long long Mt = (M + 15) / 16;
  const int Nt = (Cout + 15) / 16;
  const long long tile = (long long)blockIdx.x * 8 + wid;
  if (tile >= Mt * Nt) return;
  const long long mt = tile % Mt;        // consecutive within a block
  const int  nt = (int)(tile / Mt);      // shared within a block
  const long long m0 = mt * 16;
  const int  n0 = nt * 16;

  const int mRow = lane & 15;
  const int nCol = lane & 15;
  const int kHi  = (lane >= 16) ? 2 : 0;

  const long long m = m0 + mRow;
  const bool mOK = m < M;
  int bIdx = 0, ox = 0, oy = 0, oz = 0;
  if (mOK) {
    long long r = m;
    bIdx = (int)(r / So); r -= (long long)bIdx * So;
    ox = (int)(r / (Xo * Xo)); r -= (long long)ox * Xo * Xo;
    oy = (int)(r / Xo);
    oz = (int)(r - (long long)oy * Xo);
  }
  const int co = n0 + nCol;
  const bool nOK = co < Cout;

  v8f acc = {};
  for (int t = 0; t < 27; ++t) {
    const int dx = t / 9, dy = (t / 3) % 3, dz = t % 3;
    const int ix = ox + dx - pad, iy = oy + dy - pad, iz = oz + dz - pad;
    const bool inb = mOK && ix >= 0 && ix < Xi && iy >= 0 && iy < Xi &&
                     iz >= 0 && iz < Xi;
    const long long aBase =
        inb ? ((long long)bIdx * Cin * Si + (((long long)ix * Xi + iy) * Xi + iz)) : 0;
    const int wt = flipT ? (26 - t) : t;
    for (int ci = 0; ci < Cin; ci += 4) {
      const int k0 = ci + kHi;  // this lane's channel pair: k0, k0+1
      v2f a, bf;
      a.x = inb ? in[aBase + (long long)(k0)     * Si] : 0.0f;
      a.y = inb ? in[aBase + (long long)(k0 + 1) * Si] : 0.0f;
      if (flipT) {  // w[ci][co][flip(t)]
        bf.x = nOK ? w[((long long)(k0)     * Cout + co) * 27 + wt] : 0.0f;
        bf.y = nOK ? w[((long long)(k0 + 1) * Cout + co) * 27 + wt] : 0.0f;
      } else {      // w[co][ci][t]
        bf.x = nOK ? w[((long long)co * Cin + (k0))     * 27 + wt] : 0.0f;
        bf.y = nOK ? w[((long long)co * Cin + (k0 + 1)) * 27 + wt] : 0.0f;
      }
      acc = __builtin_amdgcn_wmma_f32_16x16x4_f32(false, a, false, bf,
                                                  (short)0, acc, false, false);
    }
  }

  if (nOK) {
    const float bval = bias[co];
#pragma unroll
    for (int r = 0; r < 8; ++r) {
      const long long mm = m0 + r + ((lane >= 16) ? 8 : 0);
      if (mm < M) {
        const long long bb = mm / So;
        const long long s  = mm - bb * So;
        out[((long long)bb * Cout + co) * So + s] = acc[r] + bval;
      }
    }
  }
}

// ---------------------------------------------------------------------------
// BatchNorm statistics (biased var, over batch+spatial) — one block / channel.
// ---------------------------------------------------------------------------
__global__ void k_bnstats(const float* __restrict__ x, float* __restrict__ mean,
                          float* __restrict__ inv, int C, long long Sp) {
  __shared__ float rs[256], rq[256];
  const int c = blockIdx.x;
  const long long n = (long long)BATCH * Sp;
  float s = 0.0f, q = 0.0f;
  for (long long i = threadIdx.x; i < n; i += blockDim.x) {
    const long long b  = i / Sp;
    const long long sp = i - b * Sp;
    const float v = x[((long long)b * C + c) * Sp + sp];
    s += v; q += v * v;
  }
  rs[threadIdx.x] = s; rq[threadIdx.x] = q;
  __syncthreads();
  for (int o = 128; o > 0; o >>= 1) {
    if (threadIdx.x < o) { rs[threadIdx.x] += rs[threadIdx.x + o];
                           rq[threadIdx.x] += rq[threadIdx.x + o]; }
    __syncthreads();
  }
  if (threadIdx.x == 0) {
    const float mu  = rs[0] / (float)n;
    const float var = rq[0] / (float)n - mu * mu;
    mean[c] = mu;
    inv[c]  = rsqrtf(var + 1e-5f);
  }
}

__global__ void k_bnrelu(float* __restrict__ x, const float* __restrict__ mean,
                         const float* __restrict__ inv, const float* __restrict__ g,
                         const float* __restrict__ beta, int C, long long Sp,
                         long long total) {
  long long i = (long long)blockIdx.x * blockDim.x + threadIdx.x;
  if (i >= total) return;
  const int c = (int)((i / Sp) % C);
  float v = (x[i] - mean[c]) * inv[c] * g[c] + beta[c];
  x[i] = v > 0.0f ? v : 0.0f;
}

// ---------------------------------------------------------------------------
// 2x2x2 maxpool with argmax (code = xoff*4 + yoff*2 + zoff, first-max wins).
// ---------------------------------------------------------------------------
__global__ void k_maxpool(const float* __restrict__ x, float* __restrict__ v,
                          int* __restrict__ idx, int C, int X, int o,
                          long long total) {
  long long i = (long long)blockIdx.x * blockDim.x + threadIdx.x;
  if (i >= total) return;
  const long long So = (long long)o * o * o;
  const long long bc = i / So;         // b*C + c
  long long r = i - bc * So;
  const int px = (int)(r / (o * o)); r -= (long long)px * o * o;
  const int py = (int)(r / o);
  const int pz = (int)(r - (long long)py * o);
  float best = -3.4e38f; int bi = 0, code = 0;
#pragma unroll
  for (int a = 0; a < 2; ++a)
#pragma unroll
    for (int bq = 0; bq < 2; ++bq)
#pragma unroll
      for (int cq = 0; cq < 2; ++cq) {
        const float val =
            x[((bc * X + (2 * px + a)) * X + (2 * py + bq)) * X + (2 * pz + cq)];
        if (val > best) { best = val; bi = code; }
        ++code;
      }
  v[i] = best; idx[i] = bi;
}

// copy skip tensor into concat buffer at channel offset
__global__ void k_copycat(const float* __restrict__ src, float* __restrict__ dst,
                          int C, long long Sp, int Ctot, int Coff, long long total) {
  long long i = (long long)blockIdx.x * blockDim.x + threadIdx.x;
  if (i >= total) return;
  const long long b = i / ((long long)C * Sp);
  const long long rem = i - b * (long long)C * Sp;
  const int c = (int)(rem / Sp);
  const long long s = rem - (long long)c * Sp;
  dst[((long long)b * Ctot + Coff + c) * Sp + s] = src[i];
}

// gather-style max-unpool into concat buffer (padded to X^3) at channel offset
__global__ void k_unpoolcat(const float* __restrict__ v, const int* __restrict__ idx,
                            float* __restrict__ dst, int C, int o, int X,
                            int Ctot, int Coff, long long total) {
  long long i = (long long)blockIdx.x * blockDim.x + threadIdx.x;
  if (i >= total) return;
  const long long Sx = (long long)X * X * X;
  const long long b = i / ((long long)C * Sx);
  long long rem = i - b * (long long)C * Sx;
  const int c = (int)(rem / Sx); rem -= (long long)c * Sx;
  const int x = (int)(rem / (X * X)); rem -= (long long)x * X * X;
  const int y = (int)(rem / X);
  const int zc = (int)(rem - (long long)y * X);
  float val = 0.0f;
  if (x < 2 * o && y < 2 * o && zc < 2 * o) {
    const int px = x >> 1, py = y >> 1, pz = zc >> 1;
    const int code = (x & 1) * 4 + (y & 1) * 2 + (zc & 1);
    const long long pi = ((((long long)b * C + c) * o + px) * o + py) * o + pz;
    if (idx[pi] == code) val = v[pi];
  }
  dst[(((long long)b * Ctot + Coff + c) * Sx) + ((long long)x * X + y) * X + zc] = val;
}

__global__ void k_sigmoid(const float* __restrict__ in, float* __restrict__ out,
                          long long n, float shift) {
  long long i = (long long)blockIdx.x * blockDim.x + threadIdx.x;
  if (i >= n) return;
  out[i] = 1.0f / (1.0f + __expf(-in[i])) + shift;
}

// ---------------------------------------------------------------------------
// occ_to_topology: per cell, 256 products over 8 corners.
// One 64-thread block per cell; corners staged in LDS.
// Thread handles 4 consecutive topologies t = tid*4+j: bits >=2 of t depend
// only on tid, so factor a common product over corners 2..7 and emit the 4
// combos of corners 0/1. Stores via float2 (topo region is 8-byte aligned).
// This kernel is the HBM roofline floor of the whole net (67 MB of output).
// ---------------------------------------------------------------------------
__global__ void k_topo(const float* __restrict__ occ, float* __restrict__ topo) {
  __shared__ float p[8];
  const int cell = blockIdx.x;            // b*32768 + linear cell
  const int b  = cell >> 15;
  const int c  = cell & 32767;
  const int cx = c >> 10, cy = (c >> 5) & 31, cz = c & 31;
  if (threadIdx.x < 8) {
    const int dxs[8] = {0, 1, 1, 0, 0, 1, 1, 0};
    const int dys[8] = {0, 0, 1, 1, 0, 0, 1, 1};
    const int dzs[8] = {0, 0, 0, 0, 1, 1, 1, 1};
    const int vtx = threadIdx.x;
    p[vtx] = occ[(((long long)b * 33 + cx + dxs[vtx]) * 33 + cy + dys[vtx]) * 33 +
                 cz + dzs[vtx]];
  }
  __syncthreads();
  const int tid = threadIdx.x;
  float common = 1.0f;
#pragma unroll
  for (int vtx = 2; vtx < 8; ++vtx) {
    const float pv = p[vtx];
    common *= ((tid >> (vtx - 2)) & 1) ? pv : (1.0f - pv);
  }
  const float p0 = p[0], p1 = p[1];
  float2 lo, hi;
  lo.x = common * (1.0f - p0) * (1.0f - p1);  // t bits (1,0) = (0,0)
  lo.y = common * p0          * (1.0f - p1);  // (0,1)
  hi.x = common * (1.0f - p0) * p1;           // (1,0)
  hi.y = common * p0          * p1;           // (1,1)
  float2* dst = (float2*)(topo + (long long)cell * 256 + tid * 4);
  dst[0] = lo;
  dst[1] = hi;
}

// ---------------------------------------------------------------------------
// Host orchestration
// ---------------------------------------------------------------------------
extern "C" void kernel_launch(void* const* d_in, const int* in_sizes, int n_in,
                              void* d_out, int out_size, void* d_ws, size_t ws_size,
                              hipStream_t stream) {
  int ii = 0;
  const float* X      = (const float*)d_in[ii++];
  const float* pn_w1  = (const float*)d_in[ii++];
  const float* pn_b1  = (const float*)d_in[ii++];
  const float* pn_w2  = (const float*)d_in[ii++];
  const float* pn_b2  = (const float*)d_in[ii++];
  struct Layer { const float *w, *b, *g, *beta; };
  auto next = [&]() { Layer L; L.w = (const float*)d_in[ii++];
    L.b = (const float*)d_in[ii++]; L.g = (const float*)d_in[ii++];
    L.beta = (const float*)d_in[ii++]; return L; };
  Layer conv1_1 = next(), conv1_2 = next(), conv2_1 = next(), conv2_2 = next(),
        conv3_1 = next(), conv3_2 = next(), conv4 = next(), deconv4 = next();
  Layer deconv3_1 = next(), deconv3_2 = next(),
        deconv2_off_1 = next(), deconv2_off_2 = next(),
        deconv2_occ_1 = next(), deconv2_occ_2 = next(),
        deconv1_off_1 = next(), deconv1_occ_1 = next();
  const float* d1off2_w = (const float*)d_in[ii++];
  const float* d1off2_b = (const float*)d_in[ii++];
  const float* d1occ2_w = (const float*)d_in[ii++];
  const float* d1occ2_b = (const float*)d_in[ii++];

  const int npts = in_sizes[0] / 3;       // 131072
  const int ptsPerB = npts / BATCH;

  // workspace bump allocator
  size_t off = 0;
  auto alloc = [&](size_t nfloats) -> float* {
    float* p = (float*)((char*)d_ws + off);
    off += nfloats * sizeof(float);
    off = (off + 255) & ~(size_t)255;
    return p;
  };
  float* meanB = alloc(128); float* invB = alloc(128);
  float* z     = alloc((size_t)npts * 16);
  float* g     = alloc((size_t)BATCH * 16 * S33);
  float* t37a  = alloc((size_t)BATCH * 16 * S37);
  float* x1    = alloc((size_t)BATCH * 16 * S37);
  float* p1v   = alloc((size_t)BATCH * 16 * S18);
  int*   p1i   = (int*)alloc((size_t)BATCH * 16 * S18);
  float* t18a  = alloc((size_t)BATCH * 32 * S18);
  float* x2    = alloc((size_t)BATCH * 32 * S18);
  float* p2v   = alloc((size_t)BATCH * 32 * S9);
  int*   p2i   = (int*)alloc((size_t)BATCH * 32 * S9);
  float* t9a   = alloc((size_t)BATCH * 64 * S9);
  float* x3    = alloc((size_t)BATCH * 64 * S9);
  float* p3v   = alloc((size_t)BATCH * 64 * S4);
  int*   p3i   = (int*)alloc((size_t)BATCH * 64 * S4);
  float* xe    = alloc((size_t)BATCH * 128 * S4);
  float* y4    = alloc((size_t)BATCH * 64 * S4);
  float* cat9  = alloc((size_t)BATCH * 128 * S9);
  float* t9b   = alloc((size_t)BATCH * 128 * S9);
  float* d32o  = alloc((size_t)BATCH * 32 * S9);
  float* cat18 = alloc((size_t)BATCH * 64 * S18);
  float* occ18a= alloc((size_t)BATCH * 64 * S18);
  float* occ18b= alloc((size_t)BATCH * 16 * S18);
  float* off18a= alloc((size_t)BATCH * 64 * S18);
  float* off18b= alloc((size_t)BATCH * 16 * S18);
  float* catOcc= alloc((size_t)BATCH * 32 * S37);
  float* catOff= alloc((size_t)BATCH * 32 * S37);
  float* occ37 = alloc((size_t)BATCH * 32 * S37);
  float* off37 = alloc((size_t)BATCH * 32 * S37);
  float* occRaw= alloc((size_t)BATCH * 1 * S33);
  float* offRaw= alloc((size_t)BATCH * 3 * S33);

  float* out_off  = (float*)d_out;                       // (2,3,33^3)
  float* out_topo = out_off + (size_t)BATCH * 3 * S33;   // (2,32768,256)
  float* out_occ  = out_topo + (size_t)BATCH * 32768 * 256;  // (2,1,33^3)

  auto grid1d = [](long long n) { return (unsigned)((n + 255) / 256); };
  auto convL = [&](const float* in, const float* w, const float* bias, float* o,
                   int Cin, int Cout, int Xi, int Xo, int pad, int flipT) {
    long long M = (long long)BATCH * Xo * Xo * Xo;
    long long tiles = ((M + 15) / 16) * ((Cout + 15) / 16);
    k_conv_wmma<<<(unsigned)((tiles + 7) / 8), 256, 0, stream>>>(
        in, w, bias, o, Cin, Cout, Xi, Xo, pad, flipT);
  };
  auto bnrelu = [&](float* buf, const Layer& L, int C, long long Sp) {
    k_bnstats<<<C, 256, 0, stream>>>(buf, meanB, invB, C, Sp);
    long long tot = (long long)BATCH * C * Sp;
    k_bnrelu<<<grid1d(tot), 256, 0, stream>>>(buf, meanB, invB, L.g, L.beta, C, Sp, tot);
  };

  // ---- PointNet + scatter to grid ----
  k_pointnet<<<(unsigned)((npts + 31) / 32), 64, 0, stream>>>(
      X, pn_w1, pn_b1, pn_w2, pn_b2, z, npts);
  k_fill0<<<grid1d((long long)BATCH * 16 * S33), 256, 0, stream>>>(
      g, (long long)BATCH * 16 * S33);
  k_scatter<<<grid1d((long long)npts * 16), 256, 0, stream>>>(X, z, g, npts, ptsPerB);

  // ---- Encoder ----
  convL(g,    conv1_1.w, conv1_1.b, t37a, 16, 16, 33, 37, 3, 0); bnrelu(t37a, conv1_1, 16, S37);
  convL(t37a, conv1_2.w, conv1_2.b, x1,   16, 16, 37, 37, 1, 0); bnrelu(x1,   conv1_2, 16, S37);
  k_maxpool<<<grid1d((long long)BATCH * 16 * S18), 256, 0, stream>>>(
      x1, p1v, p1i, 16, 37, 18, (long long)BATCH * 16 * S18);
  convL(p1v,  conv2_1.w, conv2_1.b, t18a, 16, 32, 18, 18, 1, 0); bnrelu(t18a, conv2_1, 32, S18);
  convL(t18a, conv2_2.w, conv2_2.b, x2,   32, 32, 18, 18, 1, 0); bnrelu(x2,   conv2_2, 32, S18);
  k_maxpool<<<grid1d((long long)BATCH * 32 * S9), 256, 0, stream>>>(
      x2, p2v, p2i, 32, 18, 9, (long long)BATCH * 32 * S9);
  convL(p2v,  conv3_1.w, conv3_1.b, t9a, 32, 64, 9, 9, 1, 0); bnrelu(t9a, conv3_1, 64, S9);
  convL(t9a,  conv3_2.w, conv3_2.b, x3,  64, 64, 9, 9, 1, 0); bnrelu(x3,  conv3_2, 64, S9);
  k_maxpool<<<grid1d((long long)BATCH * 64 * S4), 256, 0, stream>>>(
      x3, p3v, p3i, 64, 9, 4, (long long)BATCH * 64 * S4);
  convL(p3v, conv4.w,   conv4.b,   xe, 64, 128, 4, 4, 1, 0); bnrelu(xe, conv4,   128, S4);
  convL(xe,  deconv4.w, deconv4.b, y4, 128, 64, 4, 4, 1, 0); bnrelu(y4, deconv4,  64, S4);

  // ---- Decoder level 3 (9^3) ----
  k_copycat<<<grid1d((long long)BATCH * 64 * S9), 256, 0, stream>>>(
      x3, cat9, 64, S9, 128, 0, (long long)BATCH * 64 * S9);
  k_unpoolcat<<<grid1d((long long)BATCH * 64 * S9), 256, 0, stream>>>(
      y4, p3i, cat9, 64, 4, 9, 128, 64, (long long)BATCH * 64 * S9);
  convL(cat9, deconv3_1.w, deconv3_1.b, t9b,  128, 128, 9, 9, 1, 1); bnrelu(t9b,  deconv3_1, 128, S9);
  convL(t9b,  deconv3_2.w, deconv3_2.b, d32o, 128, 32,  9, 9, 1, 1); bnrelu(d32o, deconv3_2, 32,  S9);

  // ---- Decoder level 2 (18^3) ----
  k_copycat<<<grid1d((long long)BATCH * 32 * S18), 256, 0, stream>>>(
      x2, cat18, 32, S18, 64, 0, (long long)BATCH * 32 * S18);
  k_unpoolcat<<<grid1d((long long)BATCH * 32 * S18), 256, 0, stream>>>(
      d32o, p2i, cat18, 32, 9, 18, 64, 32, (long long)BATCH * 32 * S18);
  convL(cat18,  deconv2_occ_1.w, deconv2_occ_1.b, occ18a, 64, 64, 18, 18, 1, 1);
  bnrelu(occ18a, deconv2_occ_1, 64, S18);
  convL(occ18a, deconv2_occ_2.w, deconv2_occ_2.b, occ18b, 64, 16, 18, 18, 1, 1);
  bnrelu(occ18b, deconv2_occ_2, 16, S18);
  convL(cat18,  deconv2_off_1.w, deconv2_off_1.b, off18a, 64, 64, 18, 18, 1, 1);
  bnrelu(off18a, deconv2_off_1, 64, S18);
  convL(off18a, deconv2_off_2.w, deconv2_off_2.b, off18b, 64, 16, 18, 18, 1, 1);
  bnrelu(off18b, deconv2_off_2, 16, S18);

  // ---- Decoder level 1 (37^3) ----
  k_copycat<<<grid1d((long long)BATCH * 16 * S37), 256, 0, stream>>>(
      x1, catOcc, 16, S37, 32, 0, (long long)BATCH * 16 * S37);
  k_unpoolcat<<<grid1d((long long)BATCH * 16 * S37), 256, 0, stream>>>(
      occ18b, p1i, catOcc, 16, 18, 37, 32, 16, (long long)BATCH * 16 * S37);
  k_copycat<<<grid1d((long long)BATCH * 16 * S37), 256, 0, stream>>>(
      x1, catOff, 16, S37, 32, 0, (long long)BATCH * 16 * S37);
  k_unpoolcat<<<grid1d((long long)BATCH * 16 * S37), 256, 0, stream>>>(
      off18b, p1i, catOff, 16, 18, 37, 32, 16, (long long)BATCH * 16 * S37);
  convL(catOcc, deconv1_occ_1.w, deconv1_occ_1.b, occ37, 32, 32, 37, 37, 1, 1);
  bnrelu(occ37, deconv1_occ_1, 32, S37);
  convL(catOff, deconv1_off_1.w, deconv1_off_1.b, off37, 32, 32, 37, 37, 1, 1);
  bnrelu(off37, deconv1_off_1, 32, S37);

  // ---- Final convT (p=3 -> crop conv, pad=-1, 37 -> 33) + sigmoid ----
  convL(occ37, d1occ2_w, d1occ2_b, occRaw, 32, 1, 37, 33, -1, 1);
  convL(off37, d1off2_w, d1off2_b, offRaw, 32, 3, 37, 33, -1, 1);
  k_sigmoid<<<grid1d((long long)BATCH * 3 * S33), 256, 0, stream>>>(
      offRaw, out_off, (long long)BATCH * 3 * S33, -0.5f);
  k_sigmoid<<<grid1d((long long)BATCH * S33), 256, 0, stream>>>(
      occRaw, out_occ, (long long)BATCH * S33, 0.0f);

  // ---- Topology ----
  k_topo<<<BATCH * 32768, 64, 0, stream>>>(out_occ, out_topo);
}